// GAT_30305289241274
// MI455X (gfx1250) — compile-verified
//
#include <hip/hip_runtime.h>
#include <hip/hip_bf16.h>
#include <math.h>

typedef __attribute__((ext_vector_type(2))) float v2f;
typedef __attribute__((ext_vector_type(8))) float v8f;

#define NNODES 50000
#define NEG_SLOPE 0.2f

// ---------- helpers ----------
__device__ inline void atomicMaxF(float* addr, float val) {
  // orderable-int encoding trick; addr initialized to -inf
  if (val >= 0.0f) atomicMax((int*)addr, __float_as_int(val));
  else             atomicMin((unsigned int*)addr, __float_as_uint(val));
}

__global__ void fill_f32(float* __restrict__ p, float v, int n) {
  int i = blockIdx.x * blockDim.x + threadIdx.x;
  if (i < n) p[i] = v;
}

// ---------- fp32 WMMA GEMM: C[M x NC] = A[M x K] @ B[K x NC], row-major ----------
// One wave -> one 16x16 tile, K stepped by 4 via V_WMMA_F32_16X16X4_F32.
__global__ void gemm_wmma_f32(const float* __restrict__ A, const float* __restrict__ B,
                              float* __restrict__ C, int M, int K, int NC) {
  int wave = (int)((blockIdx.x * (size_t)blockDim.x + threadIdx.x) >> 5);
  int lane = threadIdx.x & 31;
  int tiles_n = NC >> 4;
  int tm = wave / tiles_n;
  int tn = wave - tm * tiles_n;
  if (tm >= (M >> 4)) return;          // wave-uniform exit (EXEC all-ones inside)

  int r  = lane & 15;                  // A row within tile / B col within tile
  int kh = (lane >> 4) << 1;           // 0 (lanes 0-15) or 2 (lanes 16-31)

  const float* arow = A + (size_t)(tm * 16 + r) * K;
  const float* bcol = B + (size_t)(tn * 16 + r);

  v8f acc = {};
  for (int k = 0; k < K; k += 4) {
    v2f a, b;
    a.x = arow[k + kh];
    a.y = arow[k + kh + 1];
    b.x = bcol[(size_t)(k + kh)     * NC];
    b.y = bcol[(size_t)(k + kh + 1) * NC];
    // D = A(16x4) * B(4x16) + C ; args: neg_a, A, neg_b, B, c_mod, C, reuse_a, reuse_b
    acc = __builtin_amdgcn_wmma_f32_16x16x4_f32(false, a, false, b, (short)0, acc,
                                                false, false);
  }

  // C/D layout: element (m,n): n = lane&15, m = vgpr + 8*(lane>>4)
  int mb = (lane >> 4) << 3;
  float* crow = C + (size_t)(tm * 16 + mb) * NC + tn * 16 + r;
#pragma unroll
  for (int v = 0; v < 8; ++v) crow[(size_t)v * NC] = acc[v];
}

// ---------- per-node attention coefficients: alpha[n,h] = <xl[n,h,:], a[h,:]> ----------
// C is a multiple of 4 (C=64) -> float4 streaming loads.
__global__ void attn_coef(const float* __restrict__ xl,
                          const float* __restrict__ a_src, const float* __restrict__ a_dst,
                          float* __restrict__ out_s, float* __restrict__ out_d,
                          int Nn, int H, int C) {
  int idx = blockIdx.x * blockDim.x + threadIdx.x;
  if (idx >= Nn * H) return;
  int n = idx / H, h = idx - n * H;
  const float4* row = (const float4*)(xl + (size_t)n * H * C + (size_t)h * C);
  const float4* as  = (const float4*)(a_src + (size_t)h * C);
  const float4* ad  = (const float4*)(a_dst + (size_t)h * C);
  float ss = 0.f, dd = 0.f;
  int nv = C >> 2;
  for (int c = 0; c < nv; ++c) {
    float4 v = row[c], s4 = as[c], d4 = ad[c];
    ss += v.x * s4.x + v.y * s4.y + v.z * s4.z + v.w * s4.w;
    dd += v.x * d4.x + v.y * d4.y + v.z * d4.z + v.w * d4.w;
  }
  out_s[idx] = ss;
  out_d[idx] = dd;
}

// ---------- edge pass 1: logits + segment max ----------
__global__ void edge_logit_max(const int* __restrict__ srcs, const int* __restrict__ dsts,
                               const float* __restrict__ as, const float* __restrict__ ad,
                               float* __restrict__ m, float* __restrict__ elog,
                               int E, int ET, int H) {
  int idx = blockIdx.x * blockDim.x + threadIdx.x;
  if (idx >= ET * H) return;
  int e = idx / H, h = idx - e * H;
  int s_ = (e < E) ? srcs[e] : (e - E);   // implicit self-loops for e >= E
  int d_ = (e < E) ? dsts[e] : (e - E);
  float v = as[(size_t)s_ * H + h] + ad[(size_t)d_ * H + h];
  v = (v > 0.f) ? v : NEG_SLOPE * v;      // leaky_relu
  elog[idx] = v;
  atomicMaxF(&m[(size_t)d_ * H + h], v);
}

// ---------- edge pass 2: exp(e - m[dst]) + segment sum ----------
__global__ void edge_exp_sum(const int* __restrict__ dsts,
                             float* __restrict__ elog,
                             const float* __restrict__ m, float* __restrict__ s,
                             int E, int ET, int H) {
  int idx = blockIdx.x * blockDim.x + threadIdx.x;
  if (idx >= ET * H) return;
  int e = idx / H, h = idx - e * H;
  int d_ = (e < E) ? dsts[e] : (e - E);
  float v = __expf(elog[idx] - m[(size_t)d_ * H + h]);
  elog[idx] = v;
  unsafeAtomicAdd(&s[(size_t)d_ * H + h], v);   // -> global_atomic_add_f32
}

// ---------- edge pass 3: normalize ----------
__global__ void edge_norm(const int* __restrict__ dsts,
                          float* __restrict__ elog, const float* __restrict__ s,
                          int E, int ET, int H) {
  int idx = blockIdx.x * blockDim.x + threadIdx.x;
  if (idx >= ET * H) return;
  int e = idx / H, h = idx - e * H;
  int d_ = (e < E) ? dsts[e] : (e - E);
  elog[idx] = elog[idx] / s[(size_t)d_ * H + h];
}

// ---------- edge pass 4: message aggregation (wave per edge) ----------
// float4 gather of xl[src] (GLOBAL_LOAD_B128); per-element f32 atomic scatter.
// One float4 never crosses a head boundary since C % 4 == 0.
__global__ void edge_aggregate(const int* __restrict__ srcs, const int* __restrict__ dsts,
                               const float* __restrict__ xl, const float* __restrict__ alpha,
                               float* __restrict__ out, int E, int ET, int H, int C) {
  int wave = (int)((blockIdx.x * (size_t)blockDim.x + threadIdx.x) >> 5);
  int lane = threadIdx.x & 31;
  if (wave >= ET) return;
  int s_ = (wave < E) ? srcs[wave] : (wave - E);
  int d_ = (wave < E) ? dsts[wave] : (wave - E);
  int HC = H * C;
  const float4* xr = (const float4*)(xl + (size_t)s_ * HC);
  float* orow = out + (size_t)d_ * HC;
  int nvec = HC >> 2;
  for (int i = lane; i < nvec; i += 32) {
    float4 v = xr[i];
    int base = i << 2;
    float a = alpha[(size_t)wave * H + (base / C)];
    unsafeAtomicAdd(&orow[base + 0], v.x * a);
    unsafeAtomicAdd(&orow[base + 1], v.y * a);
    unsafeAtomicAdd(&orow[base + 2], v.z * a);
    unsafeAtomicAdd(&orow[base + 3], v.w * a);
  }
}

// ---------- bias (+ optional ELU), in place ----------
__global__ void bias_act(float* __restrict__ x, const float* __restrict__ b,
                         int n, int hc, int do_elu) {
  int i = blockIdx.x * blockDim.x + threadIdx.x;
  if (i >= n) return;
  float v = x[i] + b[i % hc];
  if (do_elu) v = (v > 0.f) ? v : (__expf(v) - 1.f);
  x[i] = v;
}

// =====================================================================
extern "C" void kernel_launch(void* const* d_in, const int* in_sizes, int n_in,
                              void* d_out, int out_size, void* d_ws, size_t ws_size,
                              hipStream_t stream) {
  const float* x   = (const float*)d_in[0];
  const int*   ei  = (const int*)d_in[1];
  const float* W1  = (const float*)d_in[2];
  const float* aS1 = (const float*)d_in[3];
  const float* aD1 = (const float*)d_in[4];
  const float* b1  = (const float*)d_in[5];
  const float* W2  = (const float*)d_in[6];
  const float* aS2 = (const float*)d_in[7];
  const float* aD2 = (const float*)d_in[8];
  const float* b2  = (const float*)d_in[9];

  const int Nn = NNODES;
  const int E  = in_sizes[1] / 2;
  const int ET = E + Nn;                 // self-loops appended implicitly
  const int* src = ei;
  const int* dst = ei + E;

  // ---- carve workspace ----
  char* ws = (char*)d_ws;
  size_t off = 0;
  auto carve = [&](size_t bytes) -> char* {
    char* p = ws + off;
    off = (off + bytes + 255) & ~(size_t)255;
    return p;
  };
  float* xl1 = (float*)carve((size_t)Nn * 256 * 4);
  float* h1  = (float*)carve((size_t)Nn * 256 * 4);   // agg1 -> ELU -> layer2 input
  float* xl2 = (float*)carve((size_t)Nn * 64 * 4);
  float* as1 = (float*)carve((size_t)Nn * 4 * 4);
  float* ad1 = (float*)carve((size_t)Nn * 4 * 4);
  float* m1  = (float*)carve((size_t)Nn * 4 * 4);
  float* s1  = (float*)carve((size_t)Nn * 4 * 4);
  float* e1  = (float*)carve((size_t)ET * 4 * 4);
  float* as2 = (float*)carve((size_t)Nn * 4);
  float* ad2 = (float*)carve((size_t)Nn * 4);
  float* m2  = (float*)carve((size_t)Nn * 4);
  float* s2  = (float*)carve((size_t)Nn * 4);
  float* e2  = (float*)carve((size_t)ET * 4);
  float* outF = (float*)d_out;

  const int TB = 256;
  auto grid1 = [&](int n) { return (n + TB - 1) / TB; };

  // ================= Layer 1 (H=4, C=64) =================
  {
    int waves = (Nn / 16) * (256 / 16);
    gemm_wmma_f32<<<(waves * 32 + TB - 1) / TB, TB, 0, stream>>>(x, W1, xl1, Nn, 128, 256);
  }
  attn_coef<<<grid1(Nn * 4), TB, 0, stream>>>(xl1, aS1, aD1, as1, ad1, Nn, 4, 64);

  fill_f32<<<grid1(Nn * 4), TB, 0, stream>>>(m1, -INFINITY, Nn * 4);
  fill_f32<<<grid1(Nn * 4), TB, 0, stream>>>(s1, 0.f, Nn * 4);
  fill_f32<<<grid1(Nn * 256), TB, 0, stream>>>(h1, 0.f, Nn * 256);

  edge_logit_max<<<grid1(ET * 4), TB, 0, stream>>>(src, dst, as1, ad1, m1, e1, E, ET, 4);
  edge_exp_sum  <<<grid1(ET * 4), TB, 0, stream>>>(dst, e1, m1, s1, E, ET, 4);
  edge_norm     <<<grid1(ET * 4), TB, 0, stream>>>(dst, e1, s1, E, ET, 4);
  edge_aggregate<<<(ET * 32 + TB - 1) / TB, TB, 0, stream>>>(src, dst, xl1, e1, h1,
                                                             E, ET, 4, 64);
  bias_act<<<grid1(Nn * 256), TB, 0, stream>>>(h1, b1, Nn * 256, 256, /*elu=*/1);

  // ================= Layer 2 (H=1, C=64) =================
  {
    int waves = (Nn / 16) * (64 / 16);
    gemm_wmma_f32<<<(waves * 32 + TB - 1) / TB, TB, 0, stream>>>(h1, W2, xl2, Nn, 256, 64);
  }
  attn_coef<<<grid1(Nn), TB, 0, stream>>>(xl2, aS2, aD2, as2, ad2, Nn, 1, 64);

  fill_f32<<<grid1(Nn), TB, 0, stream>>>(m2, -INFINITY, Nn);
  fill_f32<<<grid1(Nn), TB, 0, stream>>>(s2, 0.f, Nn);
  fill_f32<<<grid1(Nn * 64), TB, 0, stream>>>(outF, 0.f, Nn * 64);

  edge_logit_max<<<grid1(ET), TB, 0, stream>>>(src, dst, as2, ad2, m2, e2, E, ET, 1);
  edge_exp_sum  <<<grid1(ET), TB, 0, stream>>>(dst, e2, m2, s2, E, ET, 1);
  edge_norm     <<<grid1(ET), TB, 0, stream>>>(dst, e2, s2, E, ET, 1);
  edge_aggregate<<<(ET * 32 + TB - 1) / TB, TB, 0, stream>>>(src, dst, xl2, e2, outF,
                                                             E, ET, 1, 64);
  bias_act<<<grid1(Nn * 64), TB, 0, stream>>>(outF, b2, Nn * 64, 64, /*elu=*/0);
}